// Handmade_conv2d_implementation_17420387352667
// MI455X (gfx1250) — compile-verified
//
#include <hip/hip_runtime.h>

typedef __attribute__((ext_vector_type(2))) float v2f;
typedef __attribute__((ext_vector_type(8))) float v8f;

namespace {
constexpr int kB     = 16;
constexpr int kCI    = 64;
constexpr int kCO    = 128;
constexpr int kH     = 128;
constexpr int kW     = 128;
constexpr int kHO    = 126;
constexpr int kWO    = 126;
constexpr int kHW    = kH * kW;          // 16384
constexpr int kOPix  = kHO * kWO;        // 15876
constexpr int kPosTot = kB * kOPix;      // 254016
constexpr int kBlkPos = 64;
constexpr int kBlkCo  = 64;
constexpr int kChunks = 36;              // K = 576 in chunks of 16 (one 3x3 tap x 16 ch)
constexpr int kRowF2  = 80;              // float2 per LDS k-pair row (160 floats; 160%64==32 -> bank-clean b64)
}

__global__ __launch_bounds__(128)
void conv3x3_implicit_gemm_wmma_f32(const float* __restrict__ data,
                                    const float* __restrict__ wgt,
                                    float* __restrict__ out)
{
    __shared__ float Asm[8 * 2 * kRowF2];   // weights tile: [k-pair][co]   as (k,k+1) float2
    __shared__ float Bsm[8 * 2 * kRowF2];   // im2col tile:  [k-pair][pos]  as (k,k+1) float2

    const int t    = threadIdx.x;
    const int lane = t & 31;
    const int ln   = lane & 15;
    const int hi   = lane >> 4;          // 0: K=k..k+1, 1: K=k+2..k+3 (WMMA 16x4 f32 layout)
    const int wv   = t >> 5;             // wave 0..3
    const int wco  = wv & 1;
    const int wpos = wv >> 1;

    const int posBlk = blockIdx.x * kBlkPos;
    const int coBlk  = blockIdx.y * kBlkCo;

    // ---- per-thread staging geometry (invariant over the whole K loop) ----
    const int sm  = t & 63;              // position (B tile) / co (A tile) within block
    const int kkb = t >> 6;              // 0/1: which element of each k-pair this thread fills

    int dbase;
    {
        const int pos = posBlk + sm;
        const int b   = pos / kOPix;
        const int rm  = pos - b * kOPix;
        const int y   = rm / kWO;
        const int x   = rm - y * kWO;
        dbase = (b * kCI * kH + y) * kW + x;     // + (ci)*kHW + r*kW + s per chunk (scalar)
    }
    const int wbase = (coBlk + sm) * (kCI * 9);  // + ci*9 + tap per chunk (scalar)

    int dIdx[8], wIdx[8], ldso[8];
#pragma unroll
    for (int j = 0; j < 8; ++j) {
        const int kk = kkb + 2 * j;              // k within chunk handled by this thread
        dIdx[j] = dbase + kk * kHW;
        wIdx[j] = wbase + kk * 9;
        ldso[j] = j * (2 * kRowF2) + sm * 2 + kkb;
    }

    v8f acc[2][2];
#pragma unroll
    for (int i = 0; i < 2; ++i)
#pragma unroll
        for (int jn = 0; jn < 2; ++jn)
            acc[i][jn] = (v8f){0.f, 0.f, 0.f, 0.f, 0.f, 0.f, 0.f, 0.f};

    for (int c = 0; c < kChunks; ++c) {
        const int rs   = c >> 2;                 // 3x3 tap 0..8
        const int ci0  = (c & 3) << 4;           // channel group 0,16,32,48
        const int r    = rs / 3;
        const int s    = rs - r * 3;
        const int soff = ci0 * kHW + r * kW + s; // scalar data offset for this chunk
        const int woff = ci0 * 9 + rs;           // scalar weight offset for this chunk

        __syncthreads();                         // prior compute done before overwrite
#pragma unroll
        for (int j = 0; j < 8; ++j) {
            Bsm[ldso[j]] = data[dIdx[j] + soff]; // coalesced along x across threads
            Asm[ldso[j]] = wgt[wIdx[j] + woff];  // tiny, cache-resident
        }
        if (c + 1 < kChunks) {                   // pull next chunk's lines toward L2/L0
            const int rs2   = (c + 1) >> 2;
            const int ci02  = ((c + 1) & 3) << 4;
            const int r2    = rs2 / 3;
            const int soff2 = ci02 * kHW + r2 * kW + (rs2 - r2 * 3);
            __builtin_prefetch(&data[dIdx[0] + soff2], 0, 0);
        }
        __syncthreads();                         // tile visible

        const v2f* Ap = (const v2f*)Asm;
        const v2f* Bp = (const v2f*)Bsm;
#pragma unroll
        for (int st = 0; st < 4; ++st) {         // K = 16 -> four 16x16x4 steps
            const int kp = (2 * st + hi) * kRowF2;
            v2f a0 = Ap[kp + wco * 32 + ln];         // weights frag, co tile 0
            v2f a1 = Ap[kp + wco * 32 + 16 + ln];    // weights frag, co tile 1
            v2f b0 = Bp[kp + wpos * 32 + ln];        // data frag, pos tile 0
            v2f b1 = Bp[kp + wpos * 32 + 16 + ln];   // data frag, pos tile 1
            acc[0][0] = __builtin_amdgcn_wmma_f32_16x16x4_f32(false, a0, false, b0,
                          (short)0, acc[0][0], false, false);
            acc[0][1] = __builtin_amdgcn_wmma_f32_16x16x4_f32(false, a0, false, b1,
                          (short)0, acc[0][1], false, false);
            acc[1][0] = __builtin_amdgcn_wmma_f32_16x16x4_f32(false, a1, false, b0,
                          (short)0, acc[1][0], false, false);
            acc[1][1] = __builtin_amdgcn_wmma_f32_16x16x4_f32(false, a1, false, b1,
                          (short)0, acc[1][1], false, false);
        }
    }

    // ---- epilogue: D rows = co (VGPR index), lanes = positions -> coalesced along x ----
#pragma unroll
    for (int jn = 0; jn < 2; ++jn) {
        const int pos = posBlk + wpos * 32 + jn * 16 + ln;
        const int b   = pos / kOPix;
        const int rm  = pos - b * kOPix;
        const int y   = rm / kWO;
        const int x   = rm - y * kWO;
        const int obase = b * (kCO * kOPix) + y * kWO + x;
#pragma unroll
        for (int i = 0; i < 2; ++i) {
            const int cb = coBlk + wco * 32 + i * 16 + hi * 8;
#pragma unroll
            for (int g = 0; g < 8; ++g) {
                out[obase + (cb + g) * kOPix] = acc[i][jn][g];
            }
        }
    }
}

extern "C" void kernel_launch(void* const* d_in, const int* in_sizes, int n_in,
                              void* d_out, int out_size, void* d_ws, size_t ws_size,
                              hipStream_t stream) {
    (void)in_sizes; (void)n_in; (void)out_size; (void)d_ws; (void)ws_size;
    const float* data = (const float*)d_in[0];
    const float* wgt  = (const float*)d_in[1];
    float* out        = (float*)d_out;
    dim3 grid(kPosTot / kBlkPos, kCO / kBlkCo, 1);   // (3969, 2)
    dim3 block(128, 1, 1);
    hipLaunchKernelGGL(conv3x3_implicit_gemm_wmma_f32, grid, block, 0, stream,
                       data, wgt, out);
}